// WaveNetModel_64020782514392
// MI455X (gfx1250) — compile-verified
//
#include <hip/hip_runtime.h>
#include <hip/hip_bf16.h>

// WaveNet inference for gfx1250 (MI455X), fp32 via V_WMMA_F32_16X16X4_F32.
// Canonical activation layout: (256 channels) x (flat time), row stride XSTR.
// dilate() collapses to: front-pad to multiple of d; conv2 -> 2-tap dilated
// conv on the flat axis; residual shift = +d; skip needs only last 4096 cols.
// Activation tiles are staged to LDS once per workgroup, via the gfx1250
// async load-to-LDS path (GLOBAL_LOAD_ASYNC_TO_LDS_B32 + s_wait_asynccnt).

#define XSTR 7168
#define NT   32   // time-columns per workgroup tile

typedef __attribute__((ext_vector_type(2))) float v2f;
typedef __attribute__((ext_vector_type(8))) float v8f;

#if __has_builtin(__builtin_amdgcn_global_load_async_to_lds_b32) && \
    __has_builtin(__builtin_amdgcn_s_wait_asynccnt)
#define USE_ASYNC_LDS 1
#else
#define USE_ASYNC_LDS 0
#endif

__device__ __forceinline__ v8f wmma4(v2f a, v2f b, v8f c) {
  // D = A(16x4) * B(4x16) + C(16x16), all fp32, wave32.
  return __builtin_amdgcn_wmma_f32_16x16x4_f32(false, a, false, b, (short)0, c,
                                               false, false);
}

__device__ __forceinline__ float sigmoidf_(float x) {
  return 1.0f / (1.0f + __expf(-x));
}

// Copy one dword global -> LDS (async path when available).
__device__ __forceinline__ void g2lds_b32(const float* g, float* l) {
#if USE_ASYNC_LDS
  __builtin_amdgcn_global_load_async_to_lds_b32((int*)g, (int*)l, 0, 0);
#else
  *l = *g;
#endif
}

__device__ __forceinline__ void g2lds_fence() {
#if USE_ASYNC_LDS
  __builtin_amdgcn_s_wait_asynccnt(0);
#endif
}

__global__ void zero_kernel(float* __restrict__ p, int n) {
  int i = blockIdx.x * 256 + threadIdx.x;
  if (i < n) p[i] = 0.0f;
}

__global__ void zero_cols_kernel(float* __restrict__ p, int pad, int stride) {
  int i = blockIdx.x * 256 + threadIdx.x;
  int total = 256 * pad;
  if (i < total) {
    int c = i / pad, j = i - c * pad;
    p[(size_t)c * stride + j] = 0.0f;
  }
}

// ---------------------------------------------------------------------------
// Fused WaveNet layer: gated dilated conv (filter,gate) + res 1x1 + shifted
// residual add + skip 1x1 accumulated into skip[512][4096] (last-4096 window).
// Xp: padded input, cols [0, T_pad). Output written at col offset pad_next.
// 8 waves; each wave owns 32 output rows of the 256-row gated conv.
// LDS (64 KB): [0,8K) floats = X tap0 tile, later reused for gated act g;
//              [8K,16K)     = X tap1 tile (also the shifted residual).
// ---------------------------------------------------------------------------
__global__ __launch_bounds__(256) void wavenet_layer_kernel(
    const float* __restrict__ Xp, int T_pad, int d, int T_out,
    const float* __restrict__ Wf, const float* __restrict__ Wg,
    const float* __restrict__ Wr, const float* __restrict__ Ws,
    float* __restrict__ Xn, int pad_next, float* __restrict__ skip) {
  __shared__ float sMem[16384];  // 64 KB
  float* sX0 = sMem;             // 256 x 32, tap 0
  float* sX1 = sMem + 8192;      // 256 x 32, tap 1 (residual source)
  float* sG  = sMem;             // 256 x 32, aliases sX0 after conv phase

  const int lane = threadIdx.x & 31;
  const int wv   = threadIdx.x >> 5;
  const int t0   = blockIdx.x * NT;
  const int ln   = lane & 15;
  const int kh   = (lane >> 4) << 1;  // K sub-offset for A/B frags: 0 or 2
  const int rowh = (lane >> 4) << 3;  // C/D row sub-offset: 0 or 8

  // ---- stage X tiles (both taps) into LDS, columns clamped at T_pad ----
  for (int idx = threadIdx.x; idx < 2 * 8192; idx += 256) {
    const int tap = idx >> 13;
    const int rem = idx & 8191;
    const int c   = rem >> 5;
    const int j   = rem & 31;
    int col = t0 + tap * d + j;
    if (col >= T_pad) col = T_pad - 1;
    g2lds_b32(Xp + (size_t)c * XSTR + col, sMem + idx);
  }
  g2lds_fence();
  __syncthreads();

  // ---- gated conv: filter & gate pre-activations, rows [wv*32, wv*32+32) ----
  v8f accF[2][2] = {};
  v8f accG[2][2] = {};
  const int mW = wv * 32;
#pragma unroll
  for (int tap = 0; tap < 2; ++tap) {
    const float* __restrict__ sXt = (tap == 0) ? sX0 : sX1;
    for (int k = 0; k < 256; k += 4) {
      const int kk = k + kh;
      v2f b[2];
      b[0].x = sXt[kk * NT + ln];
      b[0].y = sXt[(kk + 1) * NT + ln];
      b[1].x = sXt[kk * NT + 16 + ln];
      b[1].y = sXt[(kk + 1) * NT + 16 + ln];
#pragma unroll
      for (int rt = 0; rt < 2; ++rt) {
        const int m = mW + rt * 16 + ln;
        v2f af, ag;
        // filter_w / gate_w layout: (out, in, tap), tap fastest.
        af.x = Wf[((m << 8) + kk) * 2 + tap];
        af.y = Wf[((m << 8) + kk + 1) * 2 + tap];
        ag.x = Wg[((m << 8) + kk) * 2 + tap];
        ag.y = Wg[((m << 8) + kk + 1) * 2 + tap];
#pragma unroll
        for (int ct = 0; ct < 2; ++ct) {
          accF[rt][ct] = wmma4(af, b[ct], accF[rt][ct]);
          accG[rt][ct] = wmma4(ag, b[ct], accG[rt][ct]);
        }
      }
    }
  }
  __syncthreads();  // all waves done reading sX0 before sG overwrites it

  // g = tanh(filter) * sigmoid(gate) -> LDS (over the tap0 region)
#pragma unroll
  for (int rt = 0; rt < 2; ++rt)
#pragma unroll
    for (int ct = 0; ct < 2; ++ct)
#pragma unroll
      for (int e = 0; e < 8; ++e) {
        const int row = mW + rt * 16 + rowh + e;
        const int col = ct * 16 + ln;
        sG[row * NT + col] =
            tanhf(accF[rt][ct][e]) * sigmoidf_(accG[rt][ct][e]);
      }
  __syncthreads();

  // ---- residual 1x1: x_next = Wr @ g + X[:, t+d], rows [wv*32, +32) ----
  {
    v8f accR[2][2] = {};
    for (int k = 0; k < 256; k += 4) {
      const int kk = k + kh;
      v2f b0, b1;
      b0.x = sG[kk * NT + ln];
      b0.y = sG[(kk + 1) * NT + ln];
      b1.x = sG[kk * NT + 16 + ln];
      b1.y = sG[(kk + 1) * NT + 16 + ln];
#pragma unroll
      for (int rt = 0; rt < 2; ++rt) {
        const int m = mW + rt * 16 + ln;
        v2f a;
        a.x = Wr[(m << 8) + kk];
        a.y = Wr[(m << 8) + kk + 1];
        accR[rt][0] = wmma4(a, b0, accR[rt][0]);
        accR[rt][1] = wmma4(a, b1, accR[rt][1]);
      }
    }
#pragma unroll
    for (int rt = 0; rt < 2; ++rt)
#pragma unroll
      for (int ct = 0; ct < 2; ++ct)
#pragma unroll
        for (int e = 0; e < 8; ++e) {
          const int row  = mW + rt * 16 + rowh + e;
          const int col  = ct * 16 + ln;
          const int gcol = t0 + col;
          if (gcol < T_out) {
            float v = accR[rt][ct][e] + sX1[row * NT + col];
            Xn[(size_t)row * XSTR + pad_next + gcol] = v;
          }
        }
  }

  // ---- skip 1x1 (512 rows): only columns in the last-4096 window ----
  const int s0 = T_out - 4096;
  if (t0 + NT > s0) {  // uniform branch -> EXEC stays full for WMMA
    v8f accS[4][2] = {};
    const int mS = wv * 64;
    for (int k = 0; k < 256; k += 4) {
      const int kk = k + kh;
      v2f b0, b1;
      b0.x = sG[kk * NT + ln];
      b0.y = sG[(kk + 1) * NT + ln];
      b1.x = sG[kk * NT + 16 + ln];
      b1.y = sG[(kk + 1) * NT + 16 + ln];
#pragma unroll
      for (int rt = 0; rt < 4; ++rt) {
        const int m = mS + rt * 16 + ln;
        v2f a;
        a.x = Ws[(m << 8) + kk];
        a.y = Ws[(m << 8) + kk + 1];
        accS[rt][0] = wmma4(a, b0, accS[rt][0]);
        accS[rt][1] = wmma4(a, b1, accS[rt][1]);
      }
    }
#pragma unroll
    for (int rt = 0; rt < 4; ++rt)
#pragma unroll
      for (int ct = 0; ct < 2; ++ct)
#pragma unroll
        for (int e = 0; e < 8; ++e) {
          const int row  = mS + rt * 16 + rowh + e;
          const int col  = ct * 16 + ln;
          const int gcol = t0 + col;
          const int sc   = gcol - s0;
          if (sc >= 0 && gcol < T_out) {
            // exclusive ownership of (row, sc) within this launch: plain RMW
            skip[(size_t)row * 4096 + sc] += accS[rt][ct][e];
          }
        }
  }
}

// ---------------------------------------------------------------------------
// Generic 1x1 conv (GEMM): C[M x N] = act( W[M x K] @ act_in(B[K x N]) + bias )
// M = RTN * 128. Used for start conv and the two head convs.
// B tile (K x 32) staged in LDS; relu_in applied at fragment-read time.
// ---------------------------------------------------------------------------
template <int RTN, int KMAX, bool RELU_IN, bool RELU_OUT, bool HAS_BIAS>
__global__ __launch_bounds__(256) void gemm1x1_kernel(
    const float* __restrict__ W, const float* __restrict__ B,
    const float* __restrict__ bias, float* __restrict__ C, int K, int N,
    int ldb, int ldc) {
  __shared__ float sB[KMAX * NT];  // <= 64 KB

  const int lane = threadIdx.x & 31;
  const int wv   = threadIdx.x >> 5;
  const int t0   = blockIdx.x * NT;
  const int ln   = lane & 15;
  const int kh   = (lane >> 4) << 1;
  const int rowh = (lane >> 4) << 3;
  const int mW   = wv * (RTN << 4);

  for (int idx = threadIdx.x; idx < K * NT; idx += 256) {
    const int k = idx >> 5;
    const int j = idx & 31;
    int col = t0 + j;
    if (col >= N) col = N - 1;
    g2lds_b32(B + (size_t)k * ldb + col, sB + idx);
  }
  g2lds_fence();
  __syncthreads();

  v8f acc[RTN][2] = {};
  for (int k = 0; k < K; k += 4) {
    const int kk = k + kh;
    v2f b[2];
#pragma unroll
    for (int ct = 0; ct < 2; ++ct) {
      float x0 = sB[kk * NT + ct * 16 + ln];
      float x1 = sB[(kk + 1) * NT + ct * 16 + ln];
      if (RELU_IN) {
        x0 = fmaxf(x0, 0.0f);
        x1 = fmaxf(x1, 0.0f);
      }
      b[ct].x = x0;
      b[ct].y = x1;
    }
#pragma unroll
    for (int rt = 0; rt < RTN; ++rt) {
      const int m = mW + rt * 16 + ln;
      v2f a;
      a.x = W[(size_t)m * K + kk];
      a.y = W[(size_t)m * K + kk + 1];
      acc[rt][0] = wmma4(a, b[0], acc[rt][0]);
      acc[rt][1] = wmma4(a, b[1], acc[rt][1]);
    }
  }

#pragma unroll
  for (int rt = 0; rt < RTN; ++rt)
#pragma unroll
    for (int ct = 0; ct < 2; ++ct)
#pragma unroll
      for (int e = 0; e < 8; ++e) {
        const int row = mW + rt * 16 + rowh + e;
        const int col = t0 + ct * 16 + ln;
        if (col < N) {
          float v = acc[rt][ct][e];
          if (HAS_BIAS) v += bias[row];
          if (RELU_OUT) v = fmaxf(v, 0.0f);
          C[(size_t)row * ldc + col] = v;
        }
      }
}

// ---------------------------------------------------------------------------
extern "C" void kernel_launch(void* const* d_in, const int* in_sizes, int n_in,
                              void* d_out, int out_size, void* d_ws,
                              size_t ws_size, hipStream_t stream) {
  (void)in_sizes; (void)n_in; (void)out_size; (void)ws_size;

  const float* input    = (const float*)d_in[0];  // (1,256,7165)
  const float* start_w  = (const float*)d_in[1];  // (256,256)
  const float* filter_w = (const float*)d_in[2];  // (30,256,256,2)
  const float* gate_w   = (const float*)d_in[3];  // (30,256,256,2)
  const float* res_w    = (const float*)d_in[4];  // (30,256,256)
  const float* skip_w   = (const float*)d_in[5];  // (30,512,256)
  const float* end1_w   = (const float*)d_in[6];  // (512,512)
  const float* end1_b   = (const float*)d_in[7];  // (512)
  const float* end2_w   = (const float*)d_in[8];  // (256,512)
  const float* end2_b   = (const float*)d_in[9];  // (256)
  float* out = (float*)d_out;                     // (1,256,4096)

  // Workspace: 2 ping-pong activation buffers + skip + head intermediate.
  float* XA = (float*)d_ws;                       // 256 x 7168
  float* XB = XA + (size_t)256 * XSTR;            // 256 x 7168
  float* SK = XB + (size_t)256 * XSTR;            // 512 x 4096 accumulator
  float* H  = SK + (size_t)512 * 4096;            // 512 x 4096

  {  // zero the skip accumulator every call (ws is not re-poisoned to zero)
    int n = 512 * 4096;
    zero_kernel<<<(n + 255) / 256, 256, 0, stream>>>(SK, n);
  }

  // Dilation schedule: 3 blocks x (1,2,4,...,512)
  int dil[30];
  {
    int idx = 0;
    for (int b = 0; b < 3; ++b) {
      int dd = 1;
      for (int l = 0; l < 10; ++l) { dil[idx++] = dd; dd <<= 1; }
    }
  }

  // Start conv: XA = start_w @ input  (layer 0 has d=1 -> pad 0, offset 0)
  gemm1x1_kernel<2, 256, false, false, false>
      <<<(7165 + NT - 1) / NT, 256, 0, stream>>>(start_w, input, nullptr, XA,
                                                 256, 7165, 7165, XSTR);

  float* cur = XA;
  float* nxt = XB;
  int T = 7165;
  for (int i = 0; i < 30; ++i) {
    const int d    = dil[i];
    const int pad  = (d - (T % d)) % d;   // front zero-pad (flat)
    const int Tpad = T + pad;
    const int Tout = Tpad - d;
    int pad_next = 0;
    if (i < 29) {
      const int dn = dil[i + 1];
      pad_next = (dn - (Tout % dn)) % dn;
    }
    if (pad > 0) {
      int n = 256 * pad;
      zero_cols_kernel<<<(n + 255) / 256, 256, 0, stream>>>(cur, pad, XSTR);
    }
    const float* Wf = filter_w + (size_t)i * 256 * 256 * 2;
    const float* Wg = gate_w + (size_t)i * 256 * 256 * 2;
    const float* Wr = res_w + (size_t)i * 256 * 256;
    const float* Ws = skip_w + (size_t)i * 512 * 256;
    wavenet_layer_kernel<<<(Tout + NT - 1) / NT, 256, 0, stream>>>(
        cur, Tpad, d, Tout, Wf, Wg, Wr, Ws, nxt, pad_next, SK);
    T = Tout;
    float* tmp = cur; cur = nxt; nxt = tmp;
  }

  // Head: H = relu(end1 @ relu(SK) + b1);  out = end2 @ H + b2
  gemm1x1_kernel<4, 512, true, true, true><<<4096 / NT, 256, 0, stream>>>(
      end1_w, SK, end1_b, H, 512, 4096, 4096, 4096);
  gemm1x1_kernel<2, 512, false, false, true><<<4096 / NT, 256, 0, stream>>>(
      end2_w, H, end2_b, out, 512, 4096, 4096, 4096);
}